// SumStructures_14250701488277
// MI455X (gfx1250) — compile-verified
//
#include <hip/hip_runtime.h>
#include <hip/hip_bf16.h>

#define N_STRUCTURES 1000
#define ATOMS_PER_STRUCTURE 100
#define N_SAMPLES 100000
#define N_GRAD 500000
#define D_FEAT 128

typedef __attribute__((ext_vector_type(2))) float v2f;
typedef __attribute__((ext_vector_type(4))) float v4f;
typedef __attribute__((ext_vector_type(8))) float v8f;

// ---------------------------------------------------------------------------
// Zero-init of the gradient output region (d_out is poisoned by the harness;
// scatter kernel accumulates with atomics so it needs a clean slate each call).
// ---------------------------------------------------------------------------
__global__ void zero_kernel(v4f* __restrict__ p, long n4) {
    long i = (long)blockIdx.x * blockDim.x + threadIdx.x;
    long stride = (long)gridDim.x * blockDim.x;
    v4f z = {0.0f, 0.0f, 0.0f, 0.0f};
    for (; i < n4; i += stride) p[i] = z;
}

__device__ __forceinline__ int lower_bound_i32(const int* __restrict__ ids,
                                               int n, int key) {
    int lo = 0, hi = n;
    while (lo < hi) {
        int mid = (lo + hi) >> 1;
        if (ids[mid] < key) lo = mid + 1; else hi = mid;
    }
    return lo;
}

// ---------------------------------------------------------------------------
// Part 1: per-structure sum over sorted structure_ids.
// One block (8 waves) per structure; each wave owns a 16-column feature tile.
// Row reduction done with V_WMMA_F32_16X16X4_F32 where A = all-ones:
//   D[m][n] = sum_k B[k][n] + C[m][n]   (exact: ones-row dot B-columns)
// B layout (32-bit B 4x16, wave32): VGPR0 lanes0-15 = K=0, lanes16-31 = K=2;
//                                   VGPR1 lanes0-15 = K=1, lanes16-31 = K=3.
// Result: D VGPR0, lanes 0-15 hold M=0, N=lane -> the reduced 16 columns.
//
// Main loop is unguarded (no exec-mask branches) and 8-rows/iter with two
// independent accumulators so all 4 loads issue before the first wait.
// ---------------------------------------------------------------------------
__global__ void seg_sum_values_wmma(const float* __restrict__ values,
                                    const int* __restrict__ sid,
                                    float* __restrict__ out) {
    int s = blockIdx.x;
    int start = lower_bound_i32(sid, N_SAMPLES, s);
    int end   = lower_bound_i32(sid, N_SAMPLES, s + 1);

    int wave = threadIdx.x >> 5;       // 0..7 -> 16-col tile
    int lane = threadIdx.x & 31;
    int col  = wave * 16 + (lane & 15);
    int kbase = (lane < 16) ? 0 : 2;   // which K rows this lane supplies

    v8f c0 = {};                       // f32 accumulators (C/D)
    v8f c1 = {};
    v2f a; a[0] = 1.0f; a[1] = 1.0f;   // A = ones (16x4)

    // Per-lane stream pointer: rows (r0+kbase) and (r0+kbase+1) at `col`.
    const float* p = values + (long)(start + kbase) * D_FEAT + col;

    int r0 = start;
    // --- main loop: 8 rows / iter, two independent WMMA chains ------------
    for (; r0 + 8 <= end; r0 += 8) {
        v2f b0, b1;
        b0[0] = __builtin_nontemporal_load(p);
        b0[1] = __builtin_nontemporal_load(p + D_FEAT);
        b1[0] = __builtin_nontemporal_load(p + 4 * D_FEAT);
        b1[1] = __builtin_nontemporal_load(p + 5 * D_FEAT);
        c0 = __builtin_amdgcn_wmma_f32_16x16x4_f32(false, a, false, b0,
                                                   (short)0, c0, false, false);
        c1 = __builtin_amdgcn_wmma_f32_16x16x4_f32(false, a, false, b1,
                                                   (short)0, c1, false, false);
        p += 8 * D_FEAT;
    }
    // --- full 4-row step (at most one) ------------------------------------
    for (; r0 + 4 <= end; r0 += 4) {
        v2f b;
        b[0] = __builtin_nontemporal_load(p);
        b[1] = __builtin_nontemporal_load(p + D_FEAT);
        c0 = __builtin_amdgcn_wmma_f32_16x16x4_f32(false, a, false, b,
                                                   (short)0, c0, false, false);
        p += 4 * D_FEAT;
    }
    // --- guarded tail: 1..3 leftover rows (uniform branch, masked loads) --
    if (r0 < end) {
        int r = r0 + kbase;
        v2f b;
        b[0] = (r     < end) ? p[0]      : 0.0f;
        b[1] = (r + 1 < end) ? p[D_FEAT] : 0.0f;
        c0 = __builtin_amdgcn_wmma_f32_16x16x4_f32(false, a, false, b,
                                                   (short)0, c0, false, false);
    }

    if (lane < 16) out[(long)s * D_FEAT + col] = c0[0] + c1[0];
}

// ---------------------------------------------------------------------------
// Part 2: random scatter-add of 500k rows x 384 floats.
// Grid-stride over float4 quanta: b128 streaming (nontemporal) reads,
// 4x global_atomic_add_f32 to an output that is fully L2-resident (153.6 MB
// < 192 MB L2), so the RMW traffic never leaves the chip.
// ---------------------------------------------------------------------------
__global__ void scatter_grad(const v4f* __restrict__ g,
                             const int* __restrict__ gs,
                             const int* __restrict__ ga,
                             float* __restrict__ out) {
    const long total4 = (long)N_GRAD * (3 * D_FEAT / 4);   // 48M float4
    long i = (long)blockIdx.x * blockDim.x + threadIdx.x;
    long stride = (long)gridDim.x * blockDim.x;
    for (; i < total4; i += stride) {
        v4f v = __builtin_nontemporal_load(&g[i]);         // streaming read
        long row = i / 96;                                 // 96 float4 per row
        int  off = (int)(i - row * 96) * 4;
        long seg = (long)gs[row] * ATOMS_PER_STRUCTURE + (long)ga[row];
        float* dst = out + seg * (3 * D_FEAT) + off;
        atomicAdd(dst + 0, v[0]);
        atomicAdd(dst + 1, v[1]);
        atomicAdd(dst + 2, v[2]);
        atomicAdd(dst + 3, v[3]);
    }
}

extern "C" void kernel_launch(void* const* d_in, const int* in_sizes, int n_in,
                              void* d_out, int out_size, void* d_ws, size_t ws_size,
                              hipStream_t stream) {
    const float* values = (const float*)d_in[0];   // (100000, 128) f32
    const int*   sid    = (const int*)  d_in[1];   // (100000,) sorted i32
    const float* grad   = (const float*)d_in[2];   // (500000, 3, 128) f32
    const int*   gs     = (const int*)  d_in[3];   // (500000,) i32
    const int*   ga     = (const int*)  d_in[4];   // (500000,) i32

    float* out_values = (float*)d_out;                                  // 1000*128
    float* out_grad   = out_values + (long)N_STRUCTURES * D_FEAT;       // 100000*384

    // 1) zero the atomic-accumulated region (same stream -> ordered).
    long n4 = (long)N_STRUCTURES * ATOMS_PER_STRUCTURE * 3 * D_FEAT / 4;
    zero_kernel<<<2048, 256, 0, stream>>>((v4f*)out_grad, n4);

    // 2) sorted per-structure reduction via WMMA (no atomics, full overwrite).
    seg_sum_values_wmma<<<N_STRUCTURES, 256, 0, stream>>>(values, sid, out_values);

    // 3) random scatter-add, L2-resident atomics.
    scatter_grad<<<8192, 256, 0, stream>>>((const v4f*)grad, gs, ga, out_grad);
}